// SGC_63075889709455
// MI455X (gfx1250) — compile-verified
//
#include <hip/hip_runtime.h>
#include <math.h>

#define NN   128
#define CIN  64
#define TT   256
#define VV   25
#define COUT 64
#define RELC 8
#define GG   4
#define CPG  16   // channels per group (in & out for wV)
#define QKC  16   // 2*REL

typedef float v2f __attribute__((ext_vector_type(2)));
typedef float v8f __attribute__((ext_vector_type(8)));

// ---------------------------------------------------------------------------
// Kernel 1: qk[n, j, v] = bQK[j] + (1/T) * sum_t sum_ci wQK[g,og,ci]*x[n,g*16+ci,t,v]
// One block per (n, g); block produces qk channels j = g*4 .. g*4+3.
// ---------------------------------------------------------------------------
__global__ __launch_bounds__(256) void qk_kernel(
    const float* __restrict__ x, const float* __restrict__ wQK,
    const float* __restrict__ bQK, float* __restrict__ qk)
{
  __shared__ float part[CPG][16][VV + 1];
  __shared__ float xsum[CPG][VV + 1];

  const int b   = blockIdx.x;
  const int n   = b / GG, g = b % GG;
  const int tid = threadIdx.x;
  const int ci  = tid >> 4;   // 0..15 channel-in-group
  const int ts  = tid & 15;   // 0..15 t-phase

  float acc[VV];
  #pragma unroll
  for (int v = 0; v < VV; ++v) acc[v] = 0.f;

  const float* xp = x + ((size_t)n * CIN + g * CPG + ci) * TT * VV;
  for (int t = ts; t < TT; t += 16) {
    const float* row = xp + (size_t)t * VV;
    #pragma unroll
    for (int v = 0; v < VV; ++v) acc[v] += row[v];
  }
  #pragma unroll
  for (int v = 0; v < VV; ++v) part[ci][ts][v] = acc[v];
  __syncthreads();

  if (tid < CPG * VV) {
    int c = tid / VV, v = tid % VV;
    float s = 0.f;
    #pragma unroll
    for (int j = 0; j < 16; ++j) s += part[c][j][v];
    xsum[c][v] = s;
  }
  __syncthreads();

  if (tid < 4 * VV) {
    int og = tid / VV, v = tid % VV;
    float s = 0.f;
    #pragma unroll
    for (int c = 0; c < CPG; ++c)
      s += wQK[(g * 4 + og) * CPG + c] * xsum[c][v];
    qk[((size_t)n * QKC + g * 4 + og) * VV + v] = s * (1.f / TT) + bQK[g * 4 + og];
  }
}

// ---------------------------------------------------------------------------
// Kernel 2: fused v-conv + einsum, f32 WMMA (16x16x4) for both matmuls.
// One block per (n, g): 16 output channels, full T.
// LDS is aliased: phase 0 uses it for tanh matrices, main loop for vconv tile.
// Einsum B fragments (weights) are t0-invariant -> hoisted into registers.
// ---------------------------------------------------------------------------
__global__ __launch_bounds__(256) void fused_kernel(
    const float* __restrict__ x,  const float* __restrict__ Amat,
    const float* __restrict__ wV, const float* __restrict__ bV,
    const float* __restrict__ wc, const float* __restrict__ bc,
    const float* __restrict__ qk, float* __restrict__ out)
{
  // phase 0: smem[0..624] = tanh(q0[u]-k0[v]), smem[625..1249] = tanh(q1[u]-k1[v])
  // main loop: smem = vconv tile [o][s], s = t_local*25 + v  (16*400 floats)
  __shared__ float smem[CPG * 400];

  const int b    = blockIdx.x;
  const int n    = b / GG, g = b % GG;
  const int tid  = threadIdx.x;
  const int lane = tid & 31;
  const int w    = tid >> 5;      // wave id 0..7
  const int hi   = lane >> 4;     // half-wave
  const int lm   = lane & 15;
  const int kb   = hi * 2;        // K-offset for A/B fragment layout

  const float* qg = qk + (size_t)n * QKC * VV;

  for (int idx = tid; idx < 2 * VV * VV; idx += 256) {
    int j = idx / (VV * VV), r = idx % (VV * VV);
    int u = r / VV, v = r % VV;
    smem[idx] = tanhf(qg[(g * 2 + j) * VV + u] - qg[(RELC + g * 2 + j) * VV + v]);
  }
  __syncthreads();

  // ---- Hoist einsum B fragments (weights[u,v], t0-invariant) into registers.
  // Wave w owns tasks {w, w+8, w+16, w+24}; task -> (o = task/2, ut = task%2).
  v2f Bf[4][7];
  #pragma unroll
  for (int ti = 0; ti < 4; ++ti) {
    const int task = w + 8 * ti;
    const int o = task >> 1, ut = task & 1;
    const int u = ut * 16 + lm;
    const bool uok = (u < VV);
    const int uc = uok ? u : 0;                       // clamped (safe) index
    const float bco = bc[g * CPG + o];
    const float w0  = wc[(g * CPG + o) * 2 + 0];
    const float w1  = wc[(g * CPG + o) * 2 + 1];
    #pragma unroll
    for (int kc = 0; kc < 7; ++kc) {
      const int v0 = kc * 4 + kb, v1 = v0 + 1;
      const int v0c = (v0 < VV) ? v0 : 0;
      const int v1c = (v1 < VV) ? v1 : 0;
      float e0 = Amat[uc * VV + v0c] + bco + w0 * smem[uc * VV + v0c]
                                           + w1 * smem[VV * VV + uc * VV + v0c];
      float e1 = Amat[uc * VV + v1c] + bco + w0 * smem[uc * VV + v1c]
                                           + w1 * smem[VV * VV + uc * VV + v1c];
      Bf[ti][kc].x = (uok && v0 < VV) ? e0 : 0.f;
      Bf[ti][kc].y = (uok && v1 < VV) ? e1 : 0.f;
    }
  }
  __syncthreads();   // tanh reads complete; smem now becomes the vconv tile

  // wV A-fragments (M = o, K = ci): VGPR0 holds K=0/K=2, VGPR1 holds K=1/K=3
  v2f wvA[4];
  #pragma unroll
  for (int kc = 0; kc < 4; ++kc) {
    wvA[kc].x = wV[(g * CPG + lm) * CPG + kc * 4 + kb];
    wvA[kc].y = wV[(g * CPG + lm) * CPG + kc * 4 + kb + 1];
  }
  // bias C fragment: C/D layout VGPR r -> M = r + 8*hi, N = lm
  v8f cb;
  #pragma unroll
  for (int r = 0; r < 8; ++r) cb[r] = bV[g * CPG + r + 8 * hi];

  const size_t xbase = ((size_t)n * CIN + g * CPG) * TT * VV;

  for (int t0 = 0; t0 < TT; t0 += 16) {
    // ---- v-conv: 400 spatial elems = 25 s-tiles of 16; B streamed from global
    for (int st = w; st < 25; st += 8) {
      int s  = st * 16 + lm;
      int tl = s / 25, v = s % 25;
      size_t ga = xbase + (size_t)(t0 + tl) * VV + v;
      v8f d = cb;
      #pragma unroll
      for (int kc = 0; kc < 4; ++kc) {
        int ci0 = kc * 4 + kb;
        v2f bm;
        bm.x = x[ga + (size_t)ci0 * TT * VV];
        bm.y = x[ga + (size_t)(ci0 + 1) * TT * VV];
        d = __builtin_amdgcn_wmma_f32_16x16x4_f32(false, wvA[kc], false, bm,
                                                  (short)0, d, false, false);
      }
      #pragma unroll
      for (int r = 0; r < 8; ++r)
        smem[(r + 8 * hi) * 400 + st * 16 + lm] = d[r];
    }
    __syncthreads();

    // ---- einsum: out[t,u] += W[u,v] * vconv[t,v]; 4 register-resident tasks
    #pragma unroll
    for (int ti = 0; ti < 4; ++ti) {
      const int task = w + 8 * ti;
      const int o = task >> 1, ut = task & 1;
      const int u = ut * 16 + lm;
      const bool uok = (u < VV);
      const int abase = o * 400 + lm * VV;
      v8f acc = {};
      #pragma unroll
      for (int kc = 0; kc < 7; ++kc) {
        const int v0 = kc * 4 + kb, v1 = v0 + 1;
        const int v0c = (v0 < VV) ? v0 : 0;
        const int v1c = (v1 < VV) ? v1 : 0;
        v2f a;
        a.x = (v0 < VV) ? smem[abase + v0c] : 0.f;
        a.y = (v1 < VV) ? smem[abase + v1c] : 0.f;
        acc = __builtin_amdgcn_wmma_f32_16x16x4_f32(false, a, false, Bf[ti][kc],
                                                    (short)0, acc, false, false);
      }
      if (uok) {
        size_t ob = (((size_t)n * COUT + g * CPG + o) * TT + t0) * VV + u;
        #pragma unroll
        for (int r = 0; r < 8; ++r)
          out[ob + (size_t)(r + 8 * hi) * VV] = acc[r];
      }
    }
    __syncthreads();
  }
}

// ---------------------------------------------------------------------------
extern "C" void kernel_launch(void* const* d_in, const int* in_sizes, int n_in,
                              void* d_out, int out_size, void* d_ws, size_t ws_size,
                              hipStream_t stream) {
  const float* x   = (const float*)d_in[0];
  const float* A   = (const float*)d_in[1];
  const float* wQK = (const float*)d_in[2];
  const float* bQK = (const float*)d_in[3];
  const float* wV  = (const float*)d_in[4];
  const float* bV  = (const float*)d_in[5];
  const float* wc  = (const float*)d_in[6];
  const float* bc  = (const float*)d_in[7];
  float* out = (float*)d_out;
  float* qk  = (float*)d_ws;   // 128*16*25 floats = 204,800 B

  qk_kernel<<<NN * GG, 256, 0, stream>>>(x, wQK, bQK, qk);
  fused_kernel<<<NN * GG, 256, 0, stream>>>(x, A, wV, bV, wc, bc, qk, out);
}